// GraphAttentionLayer_43361989820517
// MI455X (gfx1250) — compile-verified
//
#include <hip/hip_runtime.h>

typedef float v2f __attribute__((ext_vector_type(2)));
typedef float v8f __attribute__((ext_vector_type(8)));

#define GAT_NEG_BIG (-9000000000000000.0f)
#define GAT_ALPHA   0.2f

__device__ __forceinline__ v8f wmma_f32(v2f a, v2f b, v8f c) {
    // D = A(16x4 f32) x B(4x16 f32) + C(16x16 f32), full fp32 precision
    return __builtin_amdgcn_wmma_f32_16x16x4_f32(
        /*neg_a=*/false, a, /*neg_b=*/false, b,
        /*c_mod=*/(short)0, c, /*reuse_a=*/false, /*reuse_b=*/false);
}

// ---------------------------------------------------------------------------
// Kernel 1: Wh = h @ W  (per-wave 16x64 tile via V_WMMA_F32_16X16X4_F32),
//           fsrc = Wh @ a[:64], fdst = Wh @ a[64:].
// Wh is stored PAIR-INTERLEAVED for kernel 2's WMMA-B layout:
//   WhP[b][j>>1][c][j&1]   (so (Wh[j][c], Wh[j+1][c]) is one aligned b64)
// ---------------------------------------------------------------------------
__global__ __launch_bounds__(256) void gat_wh_kernel(
    const float* __restrict__ h, const float* __restrict__ W,
    const float* __restrict__ a, float* __restrict__ WhP,
    float* __restrict__ fsrc, float* __restrict__ fdst)
{
    const int N = 2048, FIN = 128, FOUT = 64;
    const int wave = threadIdx.x >> 5;
    const int lane = threadIdx.x & 31;
    const int tile = blockIdx.x * 8 + wave;      // 0 .. B*N/16-1
    const int b    = tile >> 7;                  // tiles per batch = N/16 = 128
    const int i0   = (tile & 127) << 4;
    const int m    = lane & 15;                  // row/col-in-tile index
    const int kh   = lane >> 4;                  // half-wave selector

    v8f d0 = {}, d1 = {}, d2 = {}, d3 = {};
    const float* hrow = h + ((size_t)b * N + i0 + m) * FIN;

    #pragma unroll 8
    for (int k0 = 0; k0 < FIN; k0 += 4) {
        const int ka = k0 + 2 * kh;              // A[m][2*kh+v] per ISA layout
        v2f A = *(const v2f*)(hrow + ka);        // aligned 8B (ka even)
        const float* w0 = W + (size_t)ka * FOUT + m;        // B[2*kh+0][m + 16g]
        const float* w1 = W + (size_t)(ka + 1) * FOUT + m;  // B[2*kh+1][m + 16g]
        v2f B0; B0.x = w0[0];  B0.y = w1[0];
        v2f B1; B1.x = w0[16]; B1.y = w1[16];
        v2f B2; B2.x = w0[32]; B2.y = w1[32];
        v2f B3; B3.x = w0[48]; B3.y = w1[48];
        d0 = wmma_f32(A, B0, d0);
        d1 = wmma_f32(A, B1, d1);
        d2 = wmma_f32(A, B2, d2);
        d3 = wmma_f32(A, B3, d3);
    }

    // a_src/a_dst slices, indexed by the lane's output column m
    const float as0 = a[m],      as1 = a[16 + m], as2 = a[32 + m], as3 = a[48 + m];
    const float ad0 = a[64 + m], ad1 = a[80 + m], ad2 = a[96 + m], ad3 = a[112 + m];

    #pragma unroll
    for (int v = 0; v < 8; ++v) {
        const float e0 = d0[v], e1 = d1[v], e2 = d2[v], e3 = d3[v];
        const int row = i0 + v + 8 * kh;         // C/D layout: row = v + 8*kh
        // pair-interleaved store: WhP[b][row>>1][c][row&1]
        float* o = WhP + ((size_t)b * (N / 2) + (row >> 1)) * 128 + (row & 1);
        o[2 * m]      = e0;
        o[32 + 2 * m] = e1;
        o[64 + 2 * m] = e2;
        o[96 + 2 * m] = e3;

        float ps = e0 * as0 + e1 * as1 + e2 * as2 + e3 * as3;
        float pd = e0 * ad0 + e1 * ad1 + e2 * ad2 + e3 * ad3;
        // reduce over the 16 columns inside each half-wave (bits 0..3 only)
        #pragma unroll
        for (int off = 8; off >= 1; off >>= 1) {
            ps += __shfl_xor(ps, off, 32);
            pd += __shfl_xor(pd, off, 32);
        }
        if (m == 0) {
            fsrc[(size_t)b * N + row] = ps;
            fdst[(size_t)b * N + row] = pd;
        }
    }
}

// ---------------------------------------------------------------------------
// Kernel 2: fused masked-softmax(leaky_relu(f_i + f_j)) @ Wh, flash-style.
// One wave owns 16 output rows; streams adj in 16x16 tiles (the 268 MB adj
// is the bandwidth floor; the NxN attention matrix is never materialized).
// Online softmax rescale only runs when the row max actually changes
// (wave-uniform branch -> EXEC stays all-ones for WMMA).
// ---------------------------------------------------------------------------
__global__ __launch_bounds__(256) void gat_attn_kernel(
    const float* __restrict__ adj, const float* __restrict__ mask,
    const float* __restrict__ WhP, const float* __restrict__ fsrc,
    const float* __restrict__ fdst, float* __restrict__ out)
{
    const int N = 2048, FOUT = 64;
    const int wave = threadIdx.x >> 5;
    const int lane = threadIdx.x & 31;
    const int tile = blockIdx.x * 8 + wave;
    const int b    = tile >> 7;
    const int i0   = (tile & 127) << 4;
    const int m    = lane & 15;
    const int kh   = lane >> 4;

    const size_t bn = (size_t)b * N;
    const float fi = fsrc[bn + i0 + m];          // this lane's row logit term
    const bool  rowok = mask[bn + i0 + m] > 0.0f;
    const float* adjrow = adj + (bn + i0 + m) * N;
    const float* whb = WhP + (size_t)b * (N / 2) * 128;

    float rm = -3.0e38f;                         // running row max (A-layout: row m)
    float rs = 0.0f;                             // running row sum
    v8f acc0 = {}, acc1 = {}, acc2 = {}, acc3 = {};

    for (int j0 = 0; j0 < N; j0 += 16) {
        if (j0 + 64 < N)                         // stream-ahead on the adj row
            __builtin_prefetch(adjrow + j0 + 64, 0, 1);

        // --- logits for this lane's 8 (row m, col j) pairs, matching A layout
        float p[8];
        float cmax = -3.0e38f;
        #pragma unroll
        for (int s = 0; s < 4; ++s) {
            const int jj = j0 + 4 * s + 2 * kh;            // even
            const v2f av = *(const v2f*)(adjrow + jj);     // aligned b64
            const v2f fj = *(const v2f*)(fdst + bn + jj);  // aligned b64
            const v2f mj = *(const v2f*)(mask + bn + jj);  // aligned b64
            #pragma unroll
            for (int v = 0; v < 2; ++v) {
                const float x  = fi + fj[v];
                const float lr = (x >= 0.0f) ? x : (GAT_ALPHA * x);
                const bool  ok = (av[v] > 0.0f) && rowok && (mj[v] > 0.0f);
                const float lg = ok ? lr : GAT_NEG_BIG;
                p[2 * s + v] = lg;
                cmax = fmaxf(cmax, lg);
            }
        }
        // combine the two half-waves (they hold disjoint column subsets of row m)
        cmax = fmaxf(cmax, __shfl_xor(cmax, 16, 32));

        if (__any(cmax > rm)) {                  // wave-uniform: rescale needed
            const float nm = fmaxf(rm, cmax);
            const float sc_row = __expf(rm - nm);
            rm = nm;
            rs *= sc_row;
            // C/D row of vgpr v is (v + 8*kh) -> fetch that row's scale from
            // lane (v + 8*kh) (lanes 0..15 hold rows 0..15 in A layout)
            #pragma unroll
            for (int v = 0; v < 8; ++v) {
                const float sc = __shfl(sc_row, v + 8 * kh, 32);
                acc0[v] *= sc; acc1[v] *= sc; acc2[v] *= sc; acc3[v] *= sc;
            }
        }

        float psum = 0.0f;
        #pragma unroll
        for (int t = 0; t < 8; ++t) { p[t] = __expf(p[t] - rm); psum += p[t]; }
        psum += __shfl_xor(psum, 16, 32);
        rs += psum;

        // --- P(16x16) @ Wh(16x64) as 4 K-steps x 4 col-groups of WMMA f32.
        // B operand (Wh[jj][c], Wh[jj+1][c]) is one b64 in the interleaved layout.
        #pragma unroll
        for (int s = 0; s < 4; ++s) {
            v2f A; A.x = p[2 * s]; A.y = p[2 * s + 1];     // A[m][2*kh+v]
            const v2f* wp = (const v2f*)(whb + ((size_t)(j0 >> 1) + 2 * s + kh) * 128);
            const v2f B0 = wp[m];
            const v2f B1 = wp[16 + m];
            const v2f B2 = wp[32 + m];
            const v2f B3 = wp[48 + m];
            acc0 = wmma_f32(A, B0, acc0);
            acc1 = wmma_f32(A, B1, acc1);
            acc2 = wmma_f32(A, B2, acc2);
            acc3 = wmma_f32(A, B3, acc3);
        }
    }

    const float inv = 1.0f / rs;                 // per-row (A-layout) 1/denominator
    #pragma unroll
    for (int v = 0; v < 8; ++v) {
        const float iv = __shfl(inv, v + 8 * kh, 32);
        const int row = i0 + v + 8 * kh;
        float* o = out + (bn + row) * FOUT + m;
        o[0]  = fmaxf(acc0[v] * iv, 0.0f);
        o[16] = fmaxf(acc1[v] * iv, 0.0f);
        o[32] = fmaxf(acc2[v] * iv, 0.0f);
        o[48] = fmaxf(acc3[v] * iv, 0.0f);
    }
}

// ---------------------------------------------------------------------------
extern "C" void kernel_launch(void* const* d_in, const int* in_sizes, int n_in,
                              void* d_out, int out_size, void* d_ws, size_t ws_size,
                              hipStream_t stream) {
    (void)in_sizes; (void)n_in; (void)out_size; (void)ws_size;
    const float* h    = (const float*)d_in[0];   // (16, 2048, 128)
    const float* adj  = (const float*)d_in[1];   // (16, 2048, 2048)
    const float* mask = (const float*)d_in[2];   // (16, 2048, 1)
    const float* W    = (const float*)d_in[3];   // (128, 64)
    const float* a    = (const float*)d_in[4];   // (128, 1)
    float* out = (float*)d_out;                  // (16, 2048, 64)

    const int B = 16, N = 2048, FOUT = 64;
    float* WhP  = (float*)d_ws;                  // B*N*FOUT floats (8 MB), interleaved
    float* fsrc = WhP + (size_t)B * N * FOUT;    // B*N floats
    float* fdst = fsrc + (size_t)B * N;          // B*N floats

    const int tiles  = B * (N / 16);             // 2048 row tiles, 1 wave each
    const int blocks = tiles / 8;                // 8 waves (256 thr) per block

    gat_wh_kernel<<<blocks, 256, 0, stream>>>(h, W, a, WhP, fsrc, fdst);
    gat_attn_kernel<<<blocks, 256, 0, stream>>>(adj, mask, WhP, fsrc, fdst, out);
}